// ReduceFrames_7232724926792
// MI455X (gfx1250) — compile-verified
//
#include <hip/hip_runtime.h>
#include <hip/hip_bf16.h>

typedef __attribute__((ext_vector_type(2))) float v2f;
typedef __attribute__((ext_vector_type(8))) float v8f;

#define F_DIM   4096
#define B_DIM   32
#define C_DIM   512
#define NPAIR   2048      // F/2
#define NEWF    3073      // 1 + F*0.75

// ---------------------------------------------------------------------------
// Phase A: pred[fpair,b,o] = dot(x[2*fpair,b,:], W[o,:]) + bias[o], o in {0,1}
// Tall-skinny GEMM (65536 x 512) * (512 x 2) via V_WMMA_F32_16X16X4_F32.
// One wave = 16 rows. A: 16x4 f32 (lane<16: K=0,1 ; lane>=16: K=2,3).
// B: 4x16, columns n>=2 are zero. D: 16x16, only columns 0,1 used.
// ---------------------------------------------------------------------------
__global__ __launch_bounds__(128) void pred_gemm_kernel(
    const float* __restrict__ x, const float* __restrict__ W,
    const float* __restrict__ bias, float* __restrict__ pred) {
  const int lane  = threadIdx.x & 31;
  const int wave  = threadIdx.x >> 5;
  const int gw    = blockIdx.x * 4 + wave;      // 0..4095 (65536/16 waves)
  const int m     = lane & 15;                  // row within tile / column n
  const int khalf = (lane >> 4) << 1;           // 0 or 2

  const int R     = gw * 16 + m;                // flattened (fpair, b) row
  const int fpair = R >> 5;
  const int bb    = R & 31;
  const float* row = x + ((size_t)(2 * fpair) * B_DIM + bb) * C_DIM;

  // B-matrix source: lanes with n<2 read W[n][*], others read W[0][*]*0
  const float  bmask = (m < 2) ? 1.0f : 0.0f;
  const float* wrow  = W + (size_t)((m < 2) ? m : 0) * C_DIM;

  v8f acc = {};
  for (int k = 0; k < C_DIM; k += 4) {
    v2f a, bv;
    a.x  = row[k + khalf];
    a.y  = row[k + khalf + 1];
    bv.x = wrow[k + khalf] * bmask;
    bv.y = wrow[k + khalf + 1] * bmask;
    acc = __builtin_amdgcn_wmma_f32_16x16x4_f32(
        /*neg_a=*/false, a, /*neg_b=*/false, bv,
        /*c_mod=*/(short)0, acc, /*reuse_a=*/false, /*reuse_b=*/false);
  }

  // D layout: vgpr r -> M = r + (lane<16 ? 0 : 8); column n = lane&15
  if (m < 2) {
    const float bn    = bias[m];
    const int   mbase = (lane >> 4) * 8;
    for (int r = 0; r < 8; ++r) {
      const int Rr = gw * 16 + mbase + r;
      pred[(size_t)Rr * 2 + m] = acc[r] + bn;
    }
  }
}

// ---------------------------------------------------------------------------
// Phase B1: per-b  rstd[b] = 6 / sqrt(mean_f d^2 + 4e-20),  d = pred0-pred1.
// (score0-score1 = -d*rstd ; p0 = sigmoid(-(score1-score0)) = 1/(1+exp(d*rstd)))
// ---------------------------------------------------------------------------
__global__ __launch_bounds__(256) void stdred_kernel(
    const float* __restrict__ pred, float* __restrict__ rstd) {
  const int b = blockIdx.x, t = threadIdx.x;
  float sum = 0.0f;
  for (int p = t; p < NPAIR; p += 256) {
    const float* pr = pred + ((size_t)p * B_DIM + b) * 2;
    const float d = pr[0] - pr[1];
    sum += d * d;
  }
  __shared__ float red[256];
  red[t] = sum; __syncthreads();
  for (int off = 128; off > 0; off >>= 1) {
    if (t < off) red[t] += red[t + off];
    __syncthreads();
  }
  if (t == 0) rstd[b] = 6.0f * rsqrtf(red[0] * (1.0f / (float)NPAIR) + 4e-20f);
}

// ---------------------------------------------------------------------------
// Init inverse map: cf = -1, sep = 0   (deterministic per launch)
// ---------------------------------------------------------------------------
__global__ void init_kernel(int* __restrict__ cf, float* __restrict__ sep, int n) {
  const int i = blockIdx.x * 256 + threadIdx.x;
  if (i < n) { cf[2 * i] = -1; cf[2 * i + 1] = -1; sep[i] = 0.0f; }
}

// ---------------------------------------------------------------------------
// Phase B2: one block per b. Kept decision, block scan of `ones`,
// seq_losses, new_lengths, and the slot->frames inverse map + sep counts.
// Each thread owns 8 pairs (16 consecutive frames) so pairs never straddle.
// ---------------------------------------------------------------------------
__global__ __launch_bounds__(256) void scan_build_kernel(
    const float* __restrict__ pred, const float* __restrict__ rstd_g,
    const float* __restrict__ rand_u, const int* __restrict__ lengths,
    int* __restrict__ cf, float* __restrict__ cs, float* __restrict__ sep,
    float* __restrict__ out_tail) {
  const int b = blockIdx.x, t = threadIdx.x;
  const int len = lengths[b];
  const float rstd = rstd_g[b];

  unsigned kbits = 0, ebits = 0, obits = 0;
  float loss = 0.0f;
  int cnt = 0;
  const int p0 = t * 8;
  for (int i = 0; i < 8; ++i) {
    const int p = p0 + i;
    const float* pr = pred + ((size_t)p * B_DIM + b) * 2;
    const float q0 = pr[0], q1 = pr[1];
    const float d = q0 - q1;
    const float prob0 = 1.0f / (1.0f + expf(d * rstd));
    const float u = rand_u[(size_t)p * B_DIM + b];
    const int kept  = (prob0 > u) ? 1 : 0;
    loss += kept ? q0 : q1;
    const int fe = 2 * p, fo = 2 * p + 1;
    const int one_e = ((fe < len) ? 1 : 0) & kept;
    const int one_o = (fo < len) ? 1 : 0;
    kbits |= (unsigned)kept  << i;
    ebits |= (unsigned)one_e << i;
    obits |= (unsigned)one_o << i;
    cnt += one_e + one_o;
  }

  // block inclusive scan of cnt
  __shared__ int scn[256];
  scn[t] = cnt; __syncthreads();
  for (int off = 1; off < 256; off <<= 1) {
    int v = 0;
    if (t >= off) v = scn[t - off];
    __syncthreads();
    if (t >= off) scn[t] += v;
    __syncthreads();
  }
  const int total = scn[255];
  int c = scn[t] - cnt;   // exclusive base for this thread

  for (int i = 0; i < 8; ++i) {
    const int p = p0 + i;
    const int fe = 2 * p, fo = fe + 1;
    const int kept  = (kbits >> i) & 1;
    const int one_e = (ebits >> i) & 1;
    const int one_o = (obits >> i) & 1;
    const float scale = 0.5f * (float)(one_e + one_o);
    const int ee = c; c += one_e;
    const int eo = c; c += one_o;

    if (kept) {  // is_sep accumulation uses clamped index, no newF zeroing
      if (fe < len) atomicAdd(&sep[(size_t)min(ee, NEWF - 1) * B_DIM + b], 1.0f);
      if (fo < len) atomicAdd(&sep[(size_t)min(eo, NEWF - 1) * B_DIM + b], 1.0f);
    }
    if (scale > 0.0f) {
      if (!one_e && one_o) {
        // dropped-even pair: both frames average into slot ee (== eo)
        if (ee < NEWF) {
          const size_t s = ((size_t)ee * B_DIM + b) * 2;
          cf[s] = fe; cf[s + 1] = fo;
          cs[s] = scale; cs[s + 1] = scale;
        }
      } else {
        if (one_e && ee < NEWF) {
          const size_t s = ((size_t)ee * B_DIM + b) * 2;
          cf[s] = fe; cs[s] = scale;
        }
        if (eo < NEWF) {
          const size_t s = ((size_t)eo * B_DIM + b) * 2;
          cf[s] = fo; cs[s] = scale;
        }
      }
    }
  }

  // seq_losses reduction + new_lengths
  __shared__ float red[256];
  red[t] = loss; __syncthreads();
  for (int off = 128; off > 0; off >>= 1) {
    if (t < off) red[t] += red[t + off];
    __syncthreads();
  }
  if (t == 0) {
    out_tail[b]         = (float)min(total, NEWF);  // new_lengths
    out_tail[B_DIM + b] = red[0];                   // seq_losses
  }
}

// ---------------------------------------------------------------------------
// Phase C: gather. y[s,b,:] = sc1*x[f1,b,:] + sc2*x[f2,b,:] + sep*is_sep[:]
// 128 threads/block, float4 per thread; fully writes y (no pre-zero needed).
// ---------------------------------------------------------------------------
__global__ __launch_bounds__(128) void build_y_kernel(
    const float* __restrict__ x, const float* __restrict__ is_sep,
    const int* __restrict__ cf, const float* __restrict__ cs,
    const float* __restrict__ sep, float* __restrict__ y) {
  const int s = blockIdx.x, b = blockIdx.y, t = threadIdx.x;
  const size_t slot = (size_t)s * B_DIM + b;
  const int   f1  = cf[slot * 2],     f2  = cf[slot * 2 + 1];
  const float sc1 = cs[slot * 2],     sc2 = cs[slot * 2 + 1];
  const float sp  = sep[slot];

  const float4 r = ((const float4*)is_sep)[t];
  float a0 = r.x * sp, a1 = r.y * sp, a2 = r.z * sp, a3 = r.w * sp;
  if (f1 >= 0) {
    const float4 v = ((const float4*)(x + ((size_t)f1 * B_DIM + b) * C_DIM))[t];
    a0 += sc1 * v.x; a1 += sc1 * v.y; a2 += sc1 * v.z; a3 += sc1 * v.w;
  }
  if (f2 >= 0) {
    const float4 v = ((const float4*)(x + ((size_t)f2 * B_DIM + b) * C_DIM))[t];
    a0 += sc2 * v.x; a1 += sc2 * v.y; a2 += sc2 * v.z; a3 += sc2 * v.w;
  }
  float4 o; o.x = a0; o.y = a1; o.z = a2; o.w = a3;
  ((float4*)(y + slot * C_DIM))[t] = o;
}

// ---------------------------------------------------------------------------
extern "C" void kernel_launch(void* const* d_in, const int* in_sizes, int n_in,
                              void* d_out, int out_size, void* d_ws, size_t ws_size,
                              hipStream_t stream) {
  const float* x       = (const float*)d_in[0];
  const int*   lengths = (const int*)d_in[1];
  const float* rand_u  = (const float*)d_in[2];
  const float* W       = (const float*)d_in[3];
  const float* bias    = (const float*)d_in[4];
  const float* is_sep  = (const float*)d_in[5];
  float* y = (float*)d_out;

  // workspace layout (bytes)
  char* ws = (char*)d_ws;
  float* pred = (float*)(ws + 0);            // 65536*2 f32 = 524288 B
  float* rstd = (float*)(ws + 524288);       // 32 f32
  int*   cf   = (int*)  (ws + 524416);       // NEWF*32*2 int = 786688 B
  float* cs   = (float*)(ws + 1311104);      // NEWF*32*2 f32 = 786688 B
  float* sep  = (float*)(ws + 2097792);      // NEWF*32 f32   = 393344 B

  pred_gemm_kernel<<<1024, 128, 0, stream>>>(x, W, bias, pred);
  stdred_kernel<<<B_DIM, 256, 0, stream>>>(pred, rstd);
  init_kernel<<<(NEWF * B_DIM + 255) / 256, 256, 0, stream>>>(cf, sep, NEWF * B_DIM);
  scan_build_kernel<<<B_DIM, 256, 0, stream>>>(
      pred, rstd, rand_u, lengths, cf, cs, sep,
      y + (size_t)NEWF * B_DIM * C_DIM);
  build_y_kernel<<<dim3(NEWF, B_DIM), 128, 0, stream>>>(x, is_sep, cf, cs, sep, y);
}